// PatchLoss_30734785970526
// MI455X (gfx1250) — compile-verified
//
#include <hip/hip_runtime.h>
#include <hip/hip_bf16.h>
#include <math.h>

typedef __attribute__((ext_vector_type(2))) float v2f;
typedef __attribute__((ext_vector_type(8))) float v8f;

#define B_               64
#define H_               512
#define W_               512
#define NPATCH           4096   // (512/8)^2 patches per batch image
#define GROUPS_PER_BATCH 256    // 4096 patches / 16 patches-per-wave
#define WAVES_PER_BLOCK  8

// ---------------------------------------------------------------------------
// Stage 1: per-patch sum(|o-t|) and sum(t), 16 patches per wave via
// V_WMMA_F32_16X16X4_F32 ones-reduction (B = ones 4x16, A = patch data 16x4).
//
// Since B is all ones the WMMA output is a pure sum over K, which is
// invariant to how patch elements are assigned to (step, K-slot). Exploit
// that to use fully-coalesced b128 loads: lanes 0-15 take floats [0..3] of
// each 8-float patch row, lanes 16-31 take floats [4..7]; each float4 feeds
// two consecutive WMMA K-steps. Per load instruction the wave covers a
// contiguous 512B span.
// ---------------------------------------------------------------------------
__global__ __launch_bounds__(256) void patch_sums_wmma(
    const float* __restrict__ outp, const float* __restrict__ targ,
    float* __restrict__ lossW, float* __restrict__ meanW)
{
    const int lane = threadIdx.x & 31;
    const int wave = threadIdx.x >> 5;
    const int g    = blockIdx.x * WAVES_PER_BLOCK + wave;  // 0..16383
    const int b    = g >> 8;                               // batch
    const int gi   = g & 255;                              // patch group in batch

    const int m    = lane & 15;          // patch row M within the group
    const int half = lane >> 4;          // 0: row floats 0..3, 1: floats 4..7

    const int p  = gi * 16 + m;          // patch index within batch (same ph row)
    const int ph = p >> 6;
    const int pw = p & 63;

    const size_t base = (size_t)b * (H_ * W_) + (size_t)(ph * 8) * W_
                      + pw * 8 + half * 4;
    const float* po = outp + base;
    const float* pt = targ + base;

    v2f ones; ones.x = 1.0f; ones.y = 1.0f;   // B matrix = all ones
    v8f accL = {};                            // sum |o-t| per patch
    v8f accM = {};                            // sum t per patch

#pragma unroll
    for (int j = 0; j < 8; ++j) {             // one 8x8 patch row per iteration
        const int off = j * W_;
        float4 o = *reinterpret_cast<const float4*>(po + off);
        float4 t = *reinterpret_cast<const float4*>(pt + off);

        v2f aL0; aL0.x = fabsf(o.x - t.x); aL0.y = fabsf(o.y - t.y);
        v2f aL1; aL1.x = fabsf(o.z - t.z); aL1.y = fabsf(o.w - t.w);
        v2f aM0; aM0.x = t.x; aM0.y = t.y;
        v2f aM1; aM1.x = t.z; aM1.y = t.w;

        accL = __builtin_amdgcn_wmma_f32_16x16x4_f32(
            false, aL0, false, ones, (short)0, accL, false, false);
        accM = __builtin_amdgcn_wmma_f32_16x16x4_f32(
            false, aM0, false, ones, (short)0, accM, false, false);
        accL = __builtin_amdgcn_wmma_f32_16x16x4_f32(
            false, aL1, false, ones, (short)0, accL, false, false);
        accM = __builtin_amdgcn_wmma_f32_16x16x4_f32(
            false, aM1, false, ones, (short)0, accM, false, false);
    }

    // D column 0: lane 0 holds M=0..7 in acc[0..7], lane 16 holds M=8..15.
    if ((lane & 15) == 0) {
        const int o8 = half * 8;
        float* dl = lossW + (size_t)b * NPATCH + gi * 16 + o8;
        float* dm = meanW + (size_t)b * NPATCH + gi * 16 + o8;
        float4 l0 = make_float4(accL[0], accL[1], accL[2], accL[3]);
        float4 l1 = make_float4(accL[4], accL[5], accL[6], accL[7]);
        float4 m0 = make_float4(accM[0], accM[1], accM[2], accM[3]);
        float4 m1 = make_float4(accM[4], accM[5], accM[6], accM[7]);
        *reinterpret_cast<float4*>(dl)     = l0;
        *reinterpret_cast<float4*>(dl + 4) = l1;
        *reinterpret_cast<float4*>(dm)     = m0;
        *reinterpret_cast<float4*>(dm + 4) = m1;
    }
}

// ---------------------------------------------------------------------------
// Stage 2: per-batch argmax over 4096 patch sums for both arrays, with
// jnp.argmax tie semantics (first/smallest index wins), then agreement flag.
// ---------------------------------------------------------------------------
__global__ __launch_bounds__(256) void batch_argmax_agree(
    const float* __restrict__ lossW, const float* __restrict__ meanW,
    int* __restrict__ agree)
{
    __shared__ float sv[2][256];
    __shared__ int   si[2][256];
    const int b   = blockIdx.x;
    const int tid = threadIdx.x;
    const float* L = lossW + (size_t)b * NPATCH;
    const float* M = meanW + (size_t)b * NPATCH;

    float bl = -INFINITY; int bli = 0;
    float bm = -INFINITY; int bmi = 0;
    for (int i = tid; i < NPATCH; i += 256) {
        float l = L[i];
        if (l > bl) { bl = l; bli = i; }     // strict > keeps smallest index
        float mm = M[i];
        if (mm > bm) { bm = mm; bmi = i; }
    }
    sv[0][tid] = bl; si[0][tid] = bli;
    sv[1][tid] = bm; si[1][tid] = bmi;
    __syncthreads();

    for (int s = 128; s > 0; s >>= 1) {
        if (tid < s) {
#pragma unroll
            for (int a = 0; a < 2; ++a) {
                float v2x = sv[a][tid + s]; int i2x = si[a][tid + s];
                float v1x = sv[a][tid];     int i1x = si[a][tid];
                if (v2x > v1x || (v2x == v1x && i2x < i1x)) {
                    sv[a][tid] = v2x; si[a][tid] = i2x;
                }
            }
        }
        __syncthreads();
    }
    if (tid == 0) agree[b] = (si[0][0] == si[1][0]) ? 1 : 0;
}

// ---------------------------------------------------------------------------
// Stage 3: reduce 64 agreement flags -> (total, num_true, num_false, ratio)
// ---------------------------------------------------------------------------
__global__ void finalize_counts(const int* __restrict__ agree,
                                float* __restrict__ out)
{
    __shared__ int s[64];
    const int tid = threadIdx.x;
    s[tid] = agree[tid];
    __syncthreads();
    for (int st = 32; st > 0; st >>= 1) {
        if (tid < st) s[tid] += s[tid + st];
        __syncthreads();
    }
    if (tid == 0) {
        const int nt = s[0];
        out[0] = (float)B_;
        out[1] = (float)nt;
        out[2] = (float)(B_ - nt);
        out[3] = (float)nt / (float)B_;
    }
}

extern "C" void kernel_launch(void* const* d_in, const int* in_sizes, int n_in,
                              void* d_out, int out_size, void* d_ws, size_t ws_size,
                              hipStream_t stream)
{
    (void)in_sizes; (void)n_in; (void)out_size; (void)ws_size;
    const float* outp = (const float*)d_in[0];
    const float* targ = (const float*)d_in[1];
    // d_in[2] is patch_size (== 8), compiled in as a constant.

    float* lossW = (float*)d_ws;                           // 64*4096 floats (1 MB)
    float* meanW = lossW + (size_t)B_ * NPATCH;            // 64*4096 floats (1 MB)
    int*   agree = (int*)(meanW + (size_t)B_ * NPATCH);    // 64 ints

    const int total_groups = B_ * GROUPS_PER_BATCH;        // 16384 waves
    const int blocks = total_groups / WAVES_PER_BLOCK;     // 2048 blocks x 256 thr

    patch_sums_wmma<<<blocks, 256, 0, stream>>>(outp, targ, lossW, meanW);
    batch_argmax_agree<<<B_, 256, 0, stream>>>(lossW, meanW, agree);
    finalize_counts<<<1, 64, 0, stream>>>(agree, (float*)d_out);
}